// SSPCAB_86509231276279
// MI455X (gfx1250) — compile-verified
//
#include <hip/hip_runtime.h>

typedef __attribute__((ext_vector_type(2))) float v2f;
typedef __attribute__((ext_vector_type(8))) float v8f;
typedef __attribute__((ext_vector_type(4))) unsigned int v4u;
typedef __attribute__((ext_vector_type(4))) int v4i;
typedef __attribute__((ext_vector_type(8))) int v8i;

#define NIMG   768          // 8*96
#define IMGSZ  65536        // 256*256
#define XLO_SZ 16384        // 128*128 per image
#define Y_W    256
#define Y_IMG  32768        // 128*256 per image

// ---------------------------------------------------------------------------
// K1: fused lowpass (xlo, even-site 5x5 separable) + highpass residual xhi.
// One block = 32x32 xhi tile = 16x16 xlo tile. Wrap handled in LDS staging.
// ---------------------------------------------------------------------------
__global__ __launch_bounds__(256)
void k1_lohi(const float* __restrict__ x, const float* __restrict__ h,
             const float* __restrict__ g, float* __restrict__ xlo,
             float* __restrict__ xhi)
{
    __shared__ float xs[43 * 44];   // x patch rows [P0-6, P0+36], stride 44
    __shared__ float ls[20 * 21];   // xlo patch rows [L0-2, L0+17], stride 21
    __shared__ float hs[5], gs[7];

    const int tid = threadIdx.x;
    const int img = blockIdx.z;
    const int P0 = blockIdx.y * 32, Q0 = blockIdx.x * 32;   // xhi tile origin
    const int L0 = P0 >> 1,        M0 = Q0 >> 1;            // xlo tile origin

    if (tid < 5) hs[tid] = h[tid];
    if (tid < 7) gs[tid] = g[tid];

    const float* Xi = x + (size_t)img * IMGSZ;
    for (int idx = tid; idx < 43 * 44; idx += 256) {
        int r = idx / 44, c = idx % 44;
        if (c < 43) {
            int gr = (P0 - 6 + r + 256) & 255;
            int gc = (Q0 - 6 + c + 256) & 255;
            xs[idx] = Xi[gr * 256 + gc];
        }
    }
    __syncthreads();

    // xlo patch: 20x20, each value = 5x5 separable filter at even sites
    for (int idx = tid; idx < 400; idx += 256) {
        int rl = idx / 20, cl = idx % 20;
        float acc = 0.f;
#pragma unroll
        for (int i = 0; i < 5; ++i) {
            float t = 0.f;
#pragma unroll
            for (int j = 0; j < 5; ++j)
                t += hs[j] * xs[(2 * rl + i) * 44 + (2 * cl + j)];
            acc += hs[i] * t;
        }
        ls[rl * 21 + cl] = acc;
    }
    __syncthreads();

    // write interior 16x16 of xlo
    {
        int rl = tid >> 4, cl = tid & 15;
        xlo[(size_t)img * XLO_SZ + (L0 + rl) * 128 + (M0 + cl)] =
            ls[(rl + 2) * 21 + (cl + 2)];
    }

    // xhi = x - conv_g(upsampled xlo): only even-parity taps touch xlo
    float* XH = xhi + (size_t)img * IMGSZ;
    for (int kk = 0; kk < 4; ++kk) {
        int idx = tid + kk * 256;
        int lp = idx >> 5, lq = idx & 31;
        float s = 0.f;
#pragma unroll
        for (int i = 0; i < 7; ++i) {
            int si = lp + i - 3;
            if (si & 1) continue;                 // up row is zero
            int rl = (si >> 1) + 2;
            float t = 0.f;
#pragma unroll
            for (int j = 0; j < 7; ++j) {
                int sj = lq + j - 3;
                if (sj & 1) continue;
                int cl = (sj >> 1) + 2;
                t += gs[j] * ls[rl * 21 + cl];
            }
            s += gs[i] * t;
        }
        XH[(P0 + lp) * 256 + (Q0 + lq)] = xs[(lp + 6) * 44 + (lq + 6)] - s;
    }
}

// ---------------------------------------------------------------------------
// K2a: 7x7 conv of xhi with f0/f1 fused with quincunx row-downsample, via
// V_WMMA_F32_16X16X4_F32. Y_k[r,c] = filt2(xhi,fk)[2r,c]  (un-skewed storage).
// Banded matmul: D[m,n] += sum_t A_t[16,4] x B_t[4,16],
//   A_t[m,k] = patch[2m+i][u0+4t+k],  B_t[k,n] = fk[i, 4t+k-n].
// Band guard is pre-resolved into a zero-padded LDS table btab[k][i][j+16],
// so the hot loop issues only unpredicated LDS loads + WMMAs (no exec churn).
// ---------------------------------------------------------------------------
__global__ __launch_bounds__(128)
void k2a_qfilt(const float* __restrict__ xhi, const float* __restrict__ f0,
               const float* __restrict__ f1, float* __restrict__ Y)
{
    __shared__ float patch[37 * 72];  // xhi rows [2r0-3, 2r0+33], cols [c0-3, c0+68]
    __shared__ float btab[2 * 7 * 40]; // zero-padded band table, index j0+16 in [1,39]

    const int tid = threadIdx.x;
    const int img = blockIdx.z;
    const int r0 = blockIdx.y * 16;   // Y rows [r0, r0+16)
    const int c0 = blockIdx.x * 64;   // Y cols [c0, c0+64)

    // build padded band tables (single pass, no second barrier needed)
    for (int idx = tid; idx < 2 * 7 * 40; idx += 128) {
        int k  = idx / 280;
        int rm = idx % 280;
        int i  = rm / 40;
        int jj = rm % 40;
        int j  = jj - 16;
        float v = 0.f;
        if (j >= 0 && j < 7) v = (k == 0) ? f0[i * 7 + j] : f1[i * 7 + j];
        btab[idx] = v;
    }

    const float* Xi = xhi + (size_t)img * IMGSZ;
    for (int idx = tid; idx < 37 * 72; idx += 128) {
        int r = idx / 72, u = idx % 72;
        int gr = (2 * r0 - 3 + r + 256) & 255;
        int gc = (c0 - 3 + u + 256) & 255;
        patch[idx] = Xi[gr * 256 + gc];
    }
    __syncthreads();

    const int lane = tid & 31;
    const int wv   = tid >> 5;        // wave id 0..3
    const int n    = lane & 15;       // M for A, N for B/D
    const int kh   = lane >> 4;       // K half: lanes 16-31 hold K=2,3
    const int u0   = wv * 16;         // wave's window into the patch

    v8f acc0 = {};
    v8f acc1 = {};
#pragma unroll
    for (int i = 0; i < 7; ++i) {
#pragma unroll
        for (int t = 0; t < 6; ++t) {
            const int arow = 2 * n + i;           // quincunx: output row m -> input 2m+i
            const int acol = u0 + 4 * t + 2 * kh; // even -> 8B-aligned ds_load_b64
            v2f A;
            A.x = patch[arow * 72 + acol];
            A.y = patch[arow * 72 + acol + 1];
            const int bi = i * 40 + (4 * t + 2 * kh - n + 16); // in [1,38]
            v2f B0, B1;
            B0.x = btab[bi];
            B0.y = btab[bi + 1];
            B1.x = btab[280 + bi];
            B1.y = btab[280 + bi + 1];
            acc0 = __builtin_amdgcn_wmma_f32_16x16x4_f32(
                false, A, false, B0, (short)0, acc0, false, false);
            acc1 = __builtin_amdgcn_wmma_f32_16x16x4_f32(
                false, A, false, B1, (short)0, acc1, false, false);
        }
    }

    float* Y0 = Y + (size_t)img * Y_IMG;
    float* Y1 = Y + (size_t)NIMG * Y_IMG + (size_t)img * Y_IMG;
    const int col = c0 + u0 + n;
#pragma unroll
    for (int v = 0; v < 8; ++v) {                 // C/D layout: row = v + 8*kh
        int row = r0 + v + 8 * kh;
        Y0[row * Y_W + col] = acc0[v];
        Y1[row * Y_W + col] = acc1[v];
    }
}

// ---------------------------------------------------------------------------
// K2b: directional subbands. One block per image: the whole Y0+Y1 image pair
// (256 KB) is staged into LDS by the Tensor Data Mover (TENSOR_LOAD_TO_LDS,
// one flat 32768-element descriptor per plane, issued by wave 0, synced with
// s_wait_tensorcnt + barrier). Then the fused second 7x7 conv + quincunx
// column downsample reads only LDS:
//   sub_{k1,k2}[i,cp] = sum_{i2,j2} f_{k2}[j2,i2] *
//        Y_{k1}[(i+2cp+i2-3)%128, (6cp+2i+2i2+j2-9)%256]
// Assumption (documented): kernel has no static __shared__, so dynamic LDS
// begins at LDS byte offset 0 -> descriptor lds_addr 0 / 0x20000.
// ---------------------------------------------------------------------------

// Pack a flat-copy TDM descriptor: nelem f32 from gaddr -> LDS byte ldsoff.
// Field layout per CDNA5 ISA section 8 (D# groups 0/1; groups 2/3 zero).
// This toolchain's builtin takes 6 operands:
//   (uint32x4 g0, int32x8 g1, int32x4 g2, int32x4 g3, int32x8 extra, i32 cpol)
__device__ __forceinline__ void tdm_load_flat_f32(const float* src,
                                                  unsigned int ldsoff,
                                                  unsigned int nelem)
{
    unsigned long long ga = (unsigned long long)(uintptr_t)src;
    v4u g0;
    g0.x = 1u;                                  // count=1, is_restore=0, gather=0
    g0.y = ldsoff;                              // lds_addr (bytes)
    g0.z = (unsigned int)(ga & 0xFFFFFFFFu);    // global_addr[31:0]
    g0.w = (unsigned int)((ga >> 32) & 0x01FFFFFFu) | 0x80000000u; // addr[56:32] | type=2
    v8i g1;
    g1[0] = (int)(2u << 16);                    // workgroup_mask=0, data_size=2 (4B)
    g1[1] = (int)((nelem & 0xFFFFu) << 16);     // tensor_dim0[15:0] at bits 63:48
    g1[2] = (int)(((nelem >> 16) & 0xFFFFu)     // tensor_dim0[31:16] at bits 79:64
                  | (1u << 16));                // tensor_dim1 = 1  (bits 111:80)
    g1[3] = (int)((nelem & 0xFFFFu) << 16);     // tile_dim0 at bits 127:112
    g1[4] = 0;                                  // tile_dim1=0, tile_dim2=0 (unused)
    g1[5] = (int)nelem;                         // tensor_dim0_stride[31:0]
    g1[6] = 0;                                  // stride hi, tensor_dim1_stride lo
    g1[7] = 0;
    v4i gz4 = {0, 0, 0, 0};                     // groups 2/3 unused (<=2D tensor)
    v8i gz8 = {0, 0, 0, 0, 0, 0, 0, 0};         // trailing descriptor group unused
    __builtin_amdgcn_tensor_load_to_lds(g0, g1, gz4, gz4, gz8, 0);
}

__global__ __launch_bounds__(1024, 1)
void k2b_dfb(const float* __restrict__ Y, const float* __restrict__ f0,
             const float* __restrict__ f1, float* __restrict__ out)
{
    extern __shared__ float smem[];
    float* sY0 = smem;                 // 32768 f32 @ LDS byte 0
    float* sY1 = smem + Y_IMG;         // 32768 f32 @ LDS byte 0x20000
    float* fs  = smem + 2 * Y_IMG;     // 98 f32

    const int tid = threadIdx.x;
    const int img = blockIdx.x;

    if (tid < 49)       fs[tid] = f0[tid];
    else if (tid < 98)  fs[tid] = f1[tid - 49];

    if (tid < 32) {                    // wave 0 only: issue TDM copies
        const float* gY0 = Y + (size_t)img * Y_IMG;
        const float* gY1 = Y + (size_t)NIMG * Y_IMG + (size_t)img * Y_IMG;
        tdm_load_flat_f32(gY0, 0u, (unsigned int)Y_IMG);
        tdm_load_flat_f32(gY1, (unsigned int)(Y_IMG * 4), (unsigned int)Y_IMG);
        __builtin_amdgcn_s_wait_tensorcnt(0);
    }
    __syncthreads();

    for (int s = 0; s < 16; ++s) {
        const int flat = tid + s * 1024;          // 0..16383
        const int i  = flat >> 7;                 // output row 0..127
        const int cp = flat & 127;                // output col 0..127

        float a00 = 0.f, a01 = 0.f, a10 = 0.f, a11 = 0.f;
#pragma unroll
        for (int i2 = 0; i2 < 7; ++i2) {
            const int R  = (i + 2 * cp + i2 - 3 + 128) & 127;
            const int cb = (6 * cp + 2 * i + 2 * i2 - 9 + 256) & 255;
            const float* r0p = sY0 + R * Y_W;
            const float* r1p = sY1 + R * Y_W;
#pragma unroll
            for (int j2 = 0; j2 < 7; ++j2) {
                const int c = (cb + j2) & 255;
                float v0 = r0p[c];
                float v1 = r1p[c];
                float w0 = fs[j2 * 7 + i2];       // f0.T[i2,j2]
                float w1 = fs[49 + j2 * 7 + i2];  // f1.T[i2,j2]
                a00 += w0 * v0;
                a01 += w1 * v0;
                a10 += w0 * v1;
                a11 += w1 * v1;
            }
        }

        size_t ib = (size_t)NIMG * XLO_SZ + (size_t)img * 4 * XLO_SZ;
        out[ib + 0 * XLO_SZ + i * 128 + cp] = a00; // (y0, f0.T)
        out[ib + 1 * XLO_SZ + i * 128 + cp] = a01; // (y0, f1.T)
        out[ib + 2 * XLO_SZ + i * 128 + cp] = a10; // (y1, f0.T)
        out[ib + 3 * XLO_SZ + i * 128 + cp] = a11; // (y1, f1.T)
    }
}

// ---------------------------------------------------------------------------
extern "C" void kernel_launch(void* const* d_in, const int* in_sizes, int n_in,
                              void* d_out, int out_size, void* d_ws, size_t ws_size,
                              hipStream_t stream)
{
    (void)in_sizes; (void)n_in; (void)out_size; (void)ws_size;
    const float* x  = (const float*)d_in[0];
    const float* h  = (const float*)d_in[1];
    const float* g  = (const float*)d_in[2];
    const float* f0 = (const float*)d_in[3];
    const float* f1 = (const float*)d_in[4];
    float* out = (float*)d_out;

    float* xhi = (float*)d_ws;                             // 768*65536 f32
    float* Y   = xhi + (size_t)NIMG * IMGSZ;               // 2*768*32768 f32

    dim3 g1(8, 8, NIMG);
    k1_lohi<<<g1, 256, 0, stream>>>(x, h, g, out, xhi);

    dim3 g2(4, 8, NIMG);
    k2a_qfilt<<<g2, 128, 0, stream>>>(xhi, f0, f1, Y);

    const size_t smem_bytes = (2 * Y_IMG + 128) * sizeof(float); // 256KB + filters
    k2b_dfb<<<dim3(NIMG), 1024, smem_bytes, stream>>>(Y, f0, f1, out);
}